// GCN_75943611728724
// MI455X (gfx1250) — compile-verified
//
#include <hip/hip_runtime.h>

#define N_NODES 50000
#define N_EDGES 800000
#define F 256

typedef float v2f __attribute__((ext_vector_type(2)));
typedef float v8f __attribute__((ext_vector_type(8)));

__device__ __forceinline__ void atom_add_f32(float* p, float v) {
    // Lowers to global_atomic_add_f32 (no CAS loop) at agent scope.
    __hip_atomic_fetch_add(p, v, __ATOMIC_RELAXED, __HIP_MEMORY_SCOPE_AGENT);
}

// ---------------- degree accumulation: deg[dst] += w ----------------
__global__ void gcn_deg_kernel(const long long* __restrict__ ei,
                               const float* __restrict__ w,
                               float* __restrict__ deg) {
    int e = blockIdx.x * blockDim.x + threadIdx.x;
    if (e < N_EDGES) {
        int d = (int)ei[N_EDGES + e];
        atom_add_f32(&deg[d], w[e]);
    }
}

// ---------------- dinv_sqrt = rsqrt(deg + 1) ----------------
__global__ void gcn_dinv_kernel(const float* __restrict__ deg,
                                float* __restrict__ dinv) {
    int i = blockIdx.x * blockDim.x + threadIdx.x;
    if (i < N_NODES) dinv[i] = rsqrtf(deg[i] + 1.0f);
}

// ---------------- H = A[N x 256] @ W[256 x 256], fp32 WMMA ----------------
// One wave computes a 32x32 output tile with 4 accumulators:
// every A b64 load and B b32 load pair feeds TWO v_wmma_f32_16x16x4_f32.
// No zeroing of out-of-range A rows: D[m][*] depends only on A row m, and
// rows >= N_NODES are never stored, so a clamped address is enough.
__global__ void __launch_bounds__(128)
gcn_gemm_wmma(const float* __restrict__ A,
              const float* __restrict__ W,
              float* __restrict__ H) {
    const int lane = threadIdx.x & 31;
    const int wave = threadIdx.x >> 5;
    const int half = lane >> 4;                 // 0: K pair {0,1}; 1: {2,3}
    const int l    = lane & 15;                 // M index for A, N index for B/D
    const int m0   = blockIdx.x * 32;           // two 16-row tiles
    const int n0   = (blockIdx.y * 4 + wave) * 32; // two 16-col tiles

    const int ma  = m0 + l;
    const int mb  = m0 + 16 + l;
    const int mca = ma < N_NODES ? ma : (N_NODES - 1);  // clamp address only
    const int mcb = mb < N_NODES ? mb : (N_NODES - 1);
    const float* arow0 = A + (size_t)mca * F;
    const float* arow1 = A + (size_t)mcb * F;

    v8f c00 = {}, c01 = {}, c10 = {}, c11 = {};
    #pragma unroll 4
    for (int k = 0; k < F; k += 4) {
        const int kk = k + half * 2;
        const v2f a0 = *(const v2f*)(arow0 + kk);       // rows m0..m0+15
        const v2f a1 = *(const v2f*)(arow1 + kk);       // rows m0+16..m0+31
        const float* wp = W + (size_t)kk * F + n0 + l;
        v2f b0, b1;
        b0.x = wp[0];      b0.y = wp[F];                // cols n0..n0+15
        b1.x = wp[16];     b1.y = wp[F + 16];           // cols n0+16..n0+31
        c00 = __builtin_amdgcn_wmma_f32_16x16x4_f32(false, a0, false, b0, (short)0, c00, false, false);
        c10 = __builtin_amdgcn_wmma_f32_16x16x4_f32(false, a1, false, b0, (short)0, c10, false, false);
        c01 = __builtin_amdgcn_wmma_f32_16x16x4_f32(false, a0, false, b1, (short)0, c01, false, false);
        c11 = __builtin_amdgcn_wmma_f32_16x16x4_f32(false, a1, false, b1, (short)0, c11, false, false);
    }

    // D layout: VGPR r -> row (tile_m0 + half*8 + r), col (tile_n0 + l).
    const int r0 = m0 + half * 8;               // rows for c00/c01
    const int r1 = r0 + 16;                     // rows for c10/c11
    const int cn = n0 + l;
    if (m0 + 31 < N_NODES) {
        // Fast path: all rows valid, branch-free stores.
        #pragma unroll
        for (int r = 0; r < 8; ++r) {
            float* p0 = H + (size_t)(r0 + r) * F + cn;
            float* p1 = H + (size_t)(r1 + r) * F + cn;
            p0[0]  = c00[r];
            p0[16] = c01[r];
            p1[0]  = c10[r];
            p1[16] = c11[r];
        }
    } else {
        #pragma unroll
        for (int r = 0; r < 8; ++r) {
            if (r0 + r < N_NODES) {
                float* p0 = H + (size_t)(r0 + r) * F + cn;
                p0[0]  = c00[r];
                p0[16] = c01[r];
            }
            if (r1 + r < N_NODES) {
                float* p1 = H + (size_t)(r1 + r) * F + cn;
                p1[0]  = c10[r];
                p1[16] = c11[r];
            }
        }
    }
}

// ---------------- edge scatter: agg[dst] += norm * h[src] ----------------
// 8 edges per 256-thread block (one wave per edge). Edge metadata (src, dst,
// norm) is staged through LDS by 8 loader threads so the 32 lanes of each
// wave don't redundantly load the same i64 indices / weights / dinv gathers.
__global__ void __launch_bounds__(256)
gcn_scatter_kernel(const long long* __restrict__ ei,
                   const float* __restrict__ w,
                   const float* __restrict__ dinv,
                   const float* __restrict__ H,
                   float* agg) {
    __shared__ int   s_src[8];
    __shared__ int   s_dst[8];
    __shared__ float s_norm[8];

    const int  t    = threadIdx.x;
    const long base = (long)blockIdx.x * 8;
    if (t < 8) {
        const long e = base + t;
        int sv = -1, dv = 0;
        float nm = 0.0f;
        if (e < N_EDGES) {
            sv = (int)ei[e];
            dv = (int)ei[N_EDGES + e];
            nm = dinv[sv] * w[e] * dinv[dv];
        }
        s_src[t]  = sv;
        s_dst[t]  = dv;
        s_norm[t] = nm;
    }
    __syncthreads();

    const int wi   = t >> 5;
    const int lane = t & 31;
    const int s    = s_src[wi];
    if (s < 0) return;
    const int   d    = s_dst[wi];
    const float norm = s_norm[wi];

    const float4* hs = (const float4*)(H + (size_t)s * F) + lane * 2;
    const float4 h0 = hs[0];
    const float4 h1 = hs[1];
    float* o = agg + (size_t)d * F + lane * 8;
    atom_add_f32(o + 0, norm * h0.x);
    atom_add_f32(o + 1, norm * h0.y);
    atom_add_f32(o + 2, norm * h0.z);
    atom_add_f32(o + 3, norm * h0.w);
    atom_add_f32(o + 4, norm * h1.x);
    atom_add_f32(o + 5, norm * h1.y);
    atom_add_f32(o + 6, norm * h1.z);
    atom_add_f32(o + 7, norm * h1.w);
}

// ---------------- out = relu(agg + h * (1/deg) + b), in-place OK ----------------
__global__ void __launch_bounds__(256)
gcn_finalize_kernel(const float* agg,
                    const float* __restrict__ H,
                    const float* __restrict__ dinv,
                    const float* __restrict__ bias,
                    float* out) {
    const int i = blockIdx.x;
    const int f = threadIdx.x;
    const float di  = dinv[i];
    const float inv = di * di;                        // 1/deg
    const size_t idx = (size_t)i * F + f;
    const float v = agg[idx] + H[idx] * inv + bias[f];
    out[idx] = v > 0.0f ? v : 0.0f;
}

extern "C" void kernel_launch(void* const* d_in, const int* in_sizes, int n_in,
                              void* d_out, int out_size, void* d_ws, size_t ws_size,
                              hipStream_t stream) {
    const float*     x  = (const float*)d_in[0];
    const long long* ei = (const long long*)d_in[1];   // [2, E] int64
    const float*     w  = (const float*)d_in[2];
    const float*     W1 = (const float*)d_in[3];
    const float*     b1 = (const float*)d_in[4];
    const float*     W2 = (const float*)d_in[5];
    const float*     b2 = (const float*)d_in[6];
    float* out = (float*)d_out;

    // Workspace layout: h | agg | deg | dinv  (~103 MB, fits L2 alongside x)
    float* h    = (float*)d_ws;
    float* agg  = h + (size_t)N_NODES * F;
    float* deg  = agg + (size_t)N_NODES * F;
    float* dinv = deg + N_NODES;

    const dim3 ggrid((N_NODES + 31) / 32, 2);          // 1563 x 2 blocks, 4 waves each

    // Degree + normalization (shared by both layers).
    hipMemsetAsync(deg, 0, N_NODES * sizeof(float), stream);
    gcn_deg_kernel<<<(N_EDGES + 255) / 256, 256, 0, stream>>>(ei, w, deg);
    gcn_dinv_kernel<<<(N_NODES + 255) / 256, 256, 0, stream>>>(deg, dinv);

    // ---- Layer 1 ----
    gcn_gemm_wmma<<<ggrid, 128, 0, stream>>>(x, W1, h);
    hipMemsetAsync(agg, 0, (size_t)N_NODES * F * sizeof(float), stream);
    gcn_scatter_kernel<<<(N_EDGES + 7) / 8, 256, 0, stream>>>(ei, w, dinv, h, agg);
    gcn_finalize_kernel<<<N_NODES, 256, 0, stream>>>(agg, h, dinv, b1, agg); // y1 in agg

    // ---- Layer 2 ----
    gcn_gemm_wmma<<<ggrid, 128, 0, stream>>>(agg, W2, h);
    hipMemsetAsync(out, 0, (size_t)N_NODES * F * sizeof(float), stream);
    gcn_scatter_kernel<<<(N_EDGES + 7) / 8, 256, 0, stream>>>(ei, w, dinv, h, out);
    gcn_finalize_kernel<<<N_NODES, 256, 0, stream>>>(out, h, dinv, b2, out);
}